// Generator_18992345383023
// MI455X (gfx1250) — compile-verified
//
#include <hip/hip_runtime.h>
#include <math.h>

// ---------------------------------------------------------------------------
// DDSP-style generator for MI455X (gfx1250, wave32).
//   - GEMM-shaped stages (conv3 via im2col over a zero-padded buffer, 1x1
//     convs, osc heads) use V_WMMA_F32_16X16X4_F32, one 16x16 tile per wave.
//   - conv3 K-loop reordered (tap-outer) so there is no k->(c,d) division and
//     the padded input needs no bounds checks (no EXEC churn in inner loop).
//   - Oscillator cumsum replaced by closed-form per-256-sample-block prefix
//     (linterp is piecewise linear -> quadratic prefix) + tiny per-row scan.
//   - Noise branch collapsed analytically (only DC rfft coeff survives mask).
// ---------------------------------------------------------------------------

typedef float v2f __attribute__((ext_vector_type(2)));
typedef float v8f __attribute__((ext_vector_type(8)));

#define NBATCH   32
#define CUP      128
#define MIDC     64
#define NOSC     128
#define NSMP     16384
#define NFRM     64

static __device__ __forceinline__ float apply_act(float v, int act) {
    const float LOWF = 40.0f / 11025.0f;
    if (act == 1) { return v > 0.0f ? v : 0.2f * v; }          // leaky relu 0.2
    if (act == 2) { return fabsf(v); }                          // abs
    if (act == 3) {                                             // sigmoid^2 -> freq map
        float s = 1.0f / (1.0f + expf(-v));
        s *= s;
        return LOWF + s * (1.0f - LOWF);
    }
    return v;
}

// ---------------------------------------------------------------------------
// K0: linear  h0[b, c*4+t] = x[b,:] @ lin_w[:, c*4+t] + lin_b
// ---------------------------------------------------------------------------
__global__ void lin_kernel(const float* __restrict__ x, const float* __restrict__ w,
                           const float* __restrict__ b, float* __restrict__ out) {
    int idx = blockIdx.x * 256 + threadIdx.x;            // 32*512 = 16384
    if (idx >= NBATCH * 512) return;
    int bb = idx >> 9, j = idx & 511;
    float acc = b[j];
    for (int c = 0; c < 128; ++c) acc += x[bb * 128 + c] * w[c * 512 + j];
    out[idx] = acc;
}

// ---------------------------------------------------------------------------
// K1: band-limited 2x upsample of one (b,c) row of length n (matches
// rfft -> zero-pad -> irfft(2n) * 2).  One 64-thread block per row.
// y[m] = (1/n)(X0 + 2*sum_{k=1}^{n/2} Re(X[k] e^{i pi k m / n}))
// Output rows are written ZERO-PADDED: stride 2n+2, data at [1 .. 2n],
// zeros at [0] and [2n+1], so conv3 needs no bounds checks.
// ---------------------------------------------------------------------------
__global__ void upsample2_kernel(const float* __restrict__ in, float* __restrict__ out, int n) {
    __shared__ float sx[64], sXr[33], sXi[33];
    int row = blockIdx.x;                 // B*128 rows
    int tid = threadIdx.x;
    int nh = n >> 1;
    int ostride = 2 * n + 2;
    if (tid < n) sx[tid] = in[row * n + tid];
    __syncthreads();
    if (tid <= nh) {
        float cr = 0.0f, ci = 0.0f;
        for (int t = 0; t < n; ++t) {
            float ang = 6.28318530717958647692f * (float)(tid * t) / (float)n;
            cr += sx[t] * cosf(ang);
            ci -= sx[t] * sinf(ang);
        }
        sXr[tid] = cr; sXi[tid] = ci;
    }
    __syncthreads();
    int m = tid;
    if (m < 2 * n) {
        float y = sXr[0];
        for (int k = 1; k <= nh; ++k) {
            float th = 3.14159265358979323846f * (float)(k * m) / (float)n;
            y += 2.0f * (sXr[k] * cosf(th) - sXi[k] * sinf(th));
        }
        out[row * ostride + 1 + m] = y * (1.0f / (float)n);
    }
    if (tid == 0) {
        out[row * ostride] = 0.0f;
        out[row * ostride + ostride - 1] = 0.0f;
    }
}

// ---------------------------------------------------------------------------
// K2: conv3 ('same') as WMMA GEMM over the padded buffer.  K = 3*128,
// reordered tap-outer: for d in 0..2, for c in 0..127 step 4.
//   A[m][(d,c)] = w[m*384 + c*3 + d]            (weights (O,128,3))
//   B[(d,c)][n] = xpad[(b*128 + c)*(T+2) + t + d]   (pad already zeroed)
// One wave per 16x16 D tile.  Fragment layout (ISA 7.12.2):
//   K = vgpr + 2*(lane>=16); A-row/B-col = lane&15; D row = vgpr + 8*(lane>=16).
// ---------------------------------------------------------------------------
__global__ void conv3_wmma_kernel(const float* __restrict__ xpad, const float* __restrict__ w,
                                  const float* __restrict__ bias, float* __restrict__ out,
                                  int Cout, int T, int tshift, int act) {
    int nTileN = (NBATCH * T) >> 4;
    int nTileM = Cout >> 4;
    int gw = blockIdx.x * 8 + (threadIdx.x >> 5);
    if (gw >= nTileM * nTileN) return;
    int tm = gw / nTileN, tn = gw - tm * nTileN;

    int lane = threadIdx.x & 31;
    int half = lane >> 4;
    int l16  = lane & 15;
    int m_g  = tm * 16 + l16;                // A row (output channel)
    int n_g  = tn * 16 + l16;                // B col
    int bb   = n_g >> tshift;                // T is a power of two
    int tt   = n_g & (T - 1);
    int ps   = T + 2;                        // padded row stride

    const float* wrow = w + m_g * 384;
    const float* xcol = xpad + bb * 128 * ps + tt;
    __builtin_prefetch(wrow, 0, 3);          // global_prefetch_b8
    int cofs = half << 1;                    // this half-wave covers c = cb+cofs, +1

    v8f acc = {};
    for (int d = 0; d < 3; ++d) {
        const float* wd = wrow + d;
        const float* xd = xcol + d;
        #pragma unroll 8
        for (int cb = 0; cb < 128; cb += 4) {
            int c0 = cb + cofs;
            v2f a;  a.x  = wd[c0 * 3];       a.y  = wd[c0 * 3 + 3];
            v2f bv; bv.x = xd[c0 * ps];      bv.y = xd[c0 * ps + ps];
            acc = __builtin_amdgcn_wmma_f32_16x16x4_f32(false, a, false, bv,
                                                        (short)0, acc, false, false);
        }
    }
    for (int v = 0; v < 8; ++v) {
        int o = tm * 16 + v + 8 * half;
        float val = apply_act(acc[v] + bias[o], act);
        out[(bb * Cout + o) * T + tt] = val;
    }
}

// ---------------------------------------------------------------------------
// K3: 1x1 conv (batched GEMM M x 64 x 2048), T = 64 fixed.
// ---------------------------------------------------------------------------
__global__ void mm1x1_wmma_kernel(const float* __restrict__ xin, const float* __restrict__ w,
                                  const float* __restrict__ bias, float* __restrict__ out,
                                  int M, int act) {
    int nTileN = (NBATCH * 64) >> 4;         // 128
    int nTileM = M >> 4;
    int gw = blockIdx.x * 8 + (threadIdx.x >> 5);
    if (gw >= nTileM * nTileN) return;
    int tm = gw / nTileN, tn = gw - tm * nTileN;

    int lane = threadIdx.x & 31;
    int half = lane >> 4;
    int l16  = lane & 15;
    int m_g  = tm * 16 + l16;
    int n_g  = tn * 16 + l16;
    int bb   = n_g >> 6;
    int tt   = n_g & 63;
    const float* wrow = w + m_g * 64;
    const float* xcol = xin + bb * 4096 + tt;

    v8f acc = {};
    #pragma unroll 8
    for (int kb = 0; kb < 64; kb += 4) {
        int k0 = kb + (half << 1);
        v2f a;  a.x  = wrow[k0];             a.y  = wrow[k0 + 1];
        v2f bv; bv.x = xcol[k0 * 64];        bv.y = xcol[k0 * 64 + 64];
        acc = __builtin_amdgcn_wmma_f32_16x16x4_f32(false, a, false, bv,
                                                    (short)0, acc, false, false);
    }
    for (int v = 0; v < 8; ++v) {
        int o = tm * 16 + v + 8 * half;
        float val = apply_act(acc[v] + bias[o], act);
        out[(bb * M + o) * 64 + tt] = val;
    }
}

// ---------------------------------------------------------------------------
// K4: InstanceNorm1d over T=64 (no affine).  One 64-thread block per (b,c).
// ---------------------------------------------------------------------------
__global__ void instnorm_kernel(float* __restrict__ h) {
    __shared__ float s1[64], s2[64];
    int row = blockIdx.x;                    // 32*64
    int t = threadIdx.x;
    float v = h[row * 64 + t];
    s1[t] = v; s2[t] = v * v;
    __syncthreads();
    for (int off = 32; off > 0; off >>= 1) {
        if (t < off) { s1[t] += s1[t + off]; s2[t] += s2[t + off]; }
        __syncthreads();
    }
    float mean = s1[0] * (1.0f / 64.0f);
    float var  = s2[0] * (1.0f / 64.0f) - mean * mean;
    h[row * 64 + t] = (v - mean) * rsqrtf(var + 1e-5f);
}

// ---------------------------------------------------------------------------
// K5: fold noise head: u[m] = nm_w2[0,:] @ nm_w1[:,m];  s0 = nm_b2[0] + nm_w2[0,:].nm_b1
// ---------------------------------------------------------------------------
__global__ void noiseu_kernel(const float* __restrict__ nm_w1, const float* __restrict__ nm_b1,
                              const float* __restrict__ nm_w2, const float* __restrict__ nm_b2,
                              float* __restrict__ u, float* __restrict__ s0) {
    int m = threadIdx.x;
    float acc = 0.0f;
    for (int c = 0; c < 128; ++c) acc += nm_w2[c] * nm_w1[c * 64 + m];
    u[m] = acc;
    if (m == 0) {
        float s = nm_b2[0];
        for (int c = 0; c < 128; ++c) s += nm_w2[c] * nm_b1[c];
        s0[0] = s;
    }
}

// ---------------------------------------------------------------------------
// K6: per-frame noise value: cval[b,f] = (s0 + u . nn[b,:,f]) * sum(noise[b,f,:]) / 512.
// One wave per (b,f) pair.
// ---------------------------------------------------------------------------
__global__ void cval_kernel(const float* __restrict__ nn, const float* __restrict__ noise,
                            const float* __restrict__ u, const float* __restrict__ s0,
                            float* __restrict__ cval) {
    int pair = blockIdx.x * 8 + (threadIdx.x >> 5);   // 2048 pairs
    if (pair >= NBATCH * NFRM) return;
    int lane = threadIdx.x & 31;
    int bb = pair >> 6, f = pair & 63;
    float dc = 0.0f;
    const float* nrow = noise + (bb * NFRM + f) * 512;
    for (int j = lane; j < 512; j += 32) dc += nrow[j];
    float sp = u[lane] * nn[(bb * 64 + lane) * 64 + f]
             + u[lane + 32] * nn[(bb * 64 + lane + 32) * 64 + f];
    for (int off = 16; off > 0; off >>= 1) {
        dc += __shfl_xor(dc, off, 32);
        sp += __shfl_xor(sp, off, 32);
    }
    if (lane == 0) cval[pair] = (s0[0] + sp) * dc * (1.0f / 512.0f);
}

// ---------------------------------------------------------------------------
// K7: per-(b,osc) scan of block sums of linterp'd freq (256 samples / frame).
//   S_k = 32*fr[k-1] + 192*fr[k] + 32*fr[k+1]  (edge-clamped indices; this
//   reproduces the pos-clipping of linterp exactly: S_0 = 224f0+32f1 etc.)
// base[b,osc,k] = pi * exclusive_prefix(S)  (double accumulation).
// ---------------------------------------------------------------------------
__global__ void scan_kernel(const float* __restrict__ freq, float* __restrict__ base) {
    int gid = blockIdx.x * 256 + threadIdx.x;         // 4096 rows
    if (gid >= NBATCH * NOSC) return;
    const float* fr = freq + gid * NFRM;
    double run = 0.0;
    for (int k = 0; k < NFRM; ++k) {
        base[gid * NFRM + k] = (float)(3.14159265358979323846 * run);
        float fm = fr[k > 0 ? k - 1 : 0];
        float f0 = fr[k];
        float fp = fr[k < NFRM - 1 ? k + 1 : NFRM - 1];
        run += (double)(32.0f * fm + 192.0f * f0 + 32.0f * fp);
    }
}

// ---------------------------------------------------------------------------
// K8: oscillator-bank synthesis + noise add.  Block = (b, frame k),
// thread = sample offset o in [0,256).  Closed-form in-block prefix:
//  o<128 (left half, lo=k-1): W = 0.5(o+1) + (o+1)^2/512
//        P = fm*((o+1)-W) + f0*W ;  amp weight w = 0.5 + (o+0.5)/256
//  o>=128 (right half, lo=k): P127 = 32 fm + 96 f0; cnt = o-127;
//        V = ((o+1)^2 - 16384)/512 - cnt/2; P = P127 + f0*(cnt-V) + fp*V
//        amp weight v = (o+0.5)/256 - 0.5
// Edge frames handled by clamped neighbors (matches linterp clipping).
// ---------------------------------------------------------------------------
__global__ void synth_kernel(const float* __restrict__ amp, const float* __restrict__ freq,
                             const float* __restrict__ base, const float* __restrict__ cval,
                             float* __restrict__ out) {
    __shared__ float sFm[NOSC], sF0[NOSC], sFp[NOSC];
    __shared__ float sAm[NOSC], sA0[NOSC], sAp[NOSC];
    __shared__ float sB[NOSC];
    const float PI_F = 3.14159265358979323846f;

    int bb = blockIdx.x >> 6;
    int k  = blockIdx.x & 63;
    int o  = threadIdx.x;                               // 0..255
    int km = k > 0 ? k - 1 : 0;
    int kp = k < NFRM - 1 ? k + 1 : NFRM - 1;
    if (o < NOSC) {
        int row = (bb * NOSC + o) * NFRM;
        sFm[o] = freq[row + km]; sF0[o] = freq[row + k]; sFp[o] = freq[row + kp];
        sAm[o] = amp[row + km];  sA0[o] = amp[row + k];  sAp[o] = amp[row + kp];
        sB[o]  = base[row + k];
    }
    __syncthreads();

    float a1 = (float)(o + 1);
    bool left = (o < 128);
    float W = 0.0f, wamp = 0.0f, cnt = 0.0f, V = 0.0f, vamp = 0.0f;
    if (left) {
        W = 0.5f * a1 + a1 * a1 * (1.0f / 512.0f);
        wamp = 0.5f + ((float)o + 0.5f) * (1.0f / 256.0f);
    } else {
        cnt = (float)(o - 127);
        V = (a1 * a1 - 16384.0f) * (1.0f / 512.0f) - 0.5f * cnt;
        vamp = ((float)o + 0.5f) * (1.0f / 256.0f) - 0.5f;
    }

    float acc = 0.0f;
    for (int osc = 0; osc < NOSC; ++osc) {
        float fm = sFm[osc], f0 = sF0[osc], fp = sFp[osc];
        float P, aU;
        if (left) {
            P = fm * (a1 - W) + f0 * W;
            aU = sAm[osc] * (1.0f - wamp) + sA0[osc] * wamp;
        } else {
            float P127 = 32.0f * fm + 96.0f * f0;
            P = P127 + f0 * (cnt - V) + fp * V;
            aU = sA0[osc] * (1.0f - vamp) + sAp[osc] * vamp;
        }
        acc += sinf(sB[osc] + PI_F * P) * aU;
    }
    float nsv = cval[bb * NFRM + k] + (k > 0 ? cval[bb * NFRM + k - 1] : 0.0f);
    out[bb * NSMP + (k << 8) + o] = acc + nsv;
}

// ---------------------------------------------------------------------------
// Host side.  Workspace layout (floats); requires ~9 MB of d_ws.
// ---------------------------------------------------------------------------
extern "C" void kernel_launch(void* const* d_in, const int* in_sizes, int n_in,
                              void* d_out, int out_size, void* d_ws, size_t ws_size,
                              hipStream_t stream) {
    (void)in_sizes; (void)n_in; (void)out_size; (void)ws_size;
    const float* x      = (const float*)d_in[0];
    const float* lin_w  = (const float*)d_in[1];
    const float* lin_b  = (const float*)d_in[2];
    const float* up_w[4] = {(const float*)d_in[3], (const float*)d_in[5],
                            (const float*)d_in[7], (const float*)d_in[9]};
    const float* up_b[4] = {(const float*)d_in[4], (const float*)d_in[6],
                            (const float*)d_in[8], (const float*)d_in[10]};
    const float* fin_w  = (const float*)d_in[11];
    const float* fin_b  = (const float*)d_in[12];
    const float* hw1 = (const float*)d_in[13]; const float* hb1 = (const float*)d_in[14];
    const float* hw2 = (const float*)d_in[15]; const float* hb2 = (const float*)d_in[16];
    const float* tw1 = (const float*)d_in[17]; const float* tb1 = (const float*)d_in[18];
    const float* tw2 = (const float*)d_in[19]; const float* tb2 = (const float*)d_in[20];
    const float* amp_w = (const float*)d_in[21]; const float* amp_b = (const float*)d_in[22];
    const float* frq_w = (const float*)d_in[23]; const float* frq_b = (const float*)d_in[24];
    const float* nm_w1 = (const float*)d_in[25]; const float* nm_b1 = (const float*)d_in[26];
    const float* nm_w2 = (const float*)d_in[27]; const float* nm_b2 = (const float*)d_in[28];
    const float* noise = (const float*)d_in[29];
    float* out = (float*)d_out;

    float* W = (float*)d_ws;
    float* bufA  = W;                 // 262144  (B,128,<=64) unpadded
    float* bufU  = W + 262144;        // 270336  upsampled, padded stride T+2
    float* bufB  = W + 532480;        // 262144  unpadded
    float* hbuf  = W + 794624;        // 131072  (B,64,64)
    float* p1    = W + 925696;        // 131072
    float* hh    = W + 1056768;       // 131072
    float* q1    = W + 1187840;       // 131072
    float* nnb   = W + 1318912;       // 131072
    float* ampB  = W + 1449984;       // 262144  (B,128,64)
    float* frqB  = W + 1712128;       // 262144
    float* baseB = W + 1974272;       // 262144
    float* uvec  = W + 2236416;       // 64
    float* s0v   = W + 2236480;       // 1
    float* cvalB = W + 2236544;       // 2048    (total 2238592 floats ~ 9.0 MB)

    // front-end: linear -> (B,128,4)
    lin_kernel<<<64, 256, 0, stream>>>(x, lin_w, lin_b, bufA);

    // 4x (upsample 2x [padded] -> conv3 WMMA -> leaky)
    const float* cur = bufA;
    float* convDst[4] = {bufB, bufA, bufB, bufA};
    int T = 4, tshift = 3;
    for (int s = 0; s < 4; ++s) {
        upsample2_kernel<<<NBATCH * CUP, 64, 0, stream>>>(cur, bufU, T);
        T *= 2;
        int tiles = (CUP / 16) * ((NBATCH * T) / 16);
        conv3_wmma_kernel<<<tiles / 8, 256, 0, stream>>>(bufU, up_w[s], up_b[s],
                                                         convDst[s], CUP, T, tshift, 1);
        cur = convDst[s];
        ++tshift;
    }
    // final conv3 (128 -> 64) on the last conv output: re-pad it via a cheap
    // copy-free trick is not possible, so run the pad step with upsample's
    // writer replaced by a dedicated pad kernel?  Simpler: the last stage's
    // conv reads need padding, so pad cur into bufU (stride 66) first.
    // Reuse upsample2?  No: just a small pad-copy kernel inline below.
    {
        // pad-copy (B,128,64) -> bufU stride 66 with zero borders
        // done with lin-style kernel: one thread per padded element
        // (implemented as a lambda-free global kernel further below)
        extern __global__ void padcopy_kernel(const float*, float*);
        padcopy_kernel<<<(NBATCH * CUP * 66 + 255) / 256, 256, 0, stream>>>(cur, bufU);
        int tiles = (MIDC / 16) * ((NBATCH * 64) / 16);
        conv3_wmma_kernel<<<tiles / 8, 256, 0, stream>>>(bufU, fin_w, fin_b, hbuf,
                                                         MIDC, 64, 6, 0);
    }
    instnorm_kernel<<<NBATCH * MIDC, 64, 0, stream>>>(hbuf);

    // 1x1 conv chains (WMMA)
    mm1x1_wmma_kernel<<<64, 256, 0, stream>>>(hbuf, hw1, hb1, p1, 64, 1);
    mm1x1_wmma_kernel<<<64, 256, 0, stream>>>(p1, hw2, hb2, hh, 64, 0);
    mm1x1_wmma_kernel<<<64, 256, 0, stream>>>(hbuf, tw1, tb1, q1, 64, 1);
    mm1x1_wmma_kernel<<<64, 256, 0, stream>>>(q1, tw2, tb2, nnb, 64, 0);
    mm1x1_wmma_kernel<<<128, 256, 0, stream>>>(hh, amp_w, amp_b, ampB, 128, 2);   // |.|
    mm1x1_wmma_kernel<<<128, 256, 0, stream>>>(hh, frq_w, frq_b, frqB, 128, 3);   // sigmoid^2 map

    // noise branch (collapsed to DC) + phase base scan + synthesis
    noiseu_kernel<<<1, 64, 0, stream>>>(nm_w1, nm_b1, nm_w2, nm_b2, uvec, s0v);
    cval_kernel<<<256, 256, 0, stream>>>(nnb, noise, uvec, s0v, cvalB);
    scan_kernel<<<16, 256, 0, stream>>>(frqB, baseB);
    synth_kernel<<<NBATCH * NFRM, 256, 0, stream>>>(ampB, frqB, baseB, cvalB, out);
}

// pad-copy (B,128,64) unpadded -> padded stride 66 with zeroed borders
__global__ void padcopy_kernel(const float* __restrict__ in, float* __restrict__ outp) {
    int idx = blockIdx.x * 256 + threadIdx.x;           // B*128*66
    if (idx >= NBATCH * CUP * 66) return;
    int row = idx / 66;
    int p   = idx - row * 66;
    outp[idx] = (p >= 1 && p <= 64) ? in[row * 64 + (p - 1)] : 0.0f;
}